// GCNEncoder2_52716428591749
// MI455X (gfx1250) — compile-verified
//
#include <hip/hip_runtime.h>

typedef float v2f __attribute__((ext_vector_type(2)));
typedef float v8f __attribute__((ext_vector_type(8)));

// ---------------------------------------------------------------------------
// Dense GEMM: Y[r,c] = act( X[r,:] @ W[:,:] + bias[c] ), K = Ncols = 64.
// One wave -> 16 rows x 64 cols via V_WMMA_F32_16X16X4_F32 (fp32 matrix pipe).
// W is staged TRANSPOSED in LDS (Wt[c][k]) so each B fragment
// (K = k0+2*hi, k0+2*hi+1 ; N = c) is one aligned ds_load_b64 into an even
// VGPR pair -- no register shuffling in front of the WMMA.
// Fragment layouts per CDNA5 ISA 7.12.2:
//   A 16x4 : lanes 0-15 {K0,K1}, lanes 16-31 {K2,K3}, M = lane%16
//   B 4x16 : lanes 0-15 {K0,K1}, lanes 16-31 {K2,K3}, N = lane%16
//   C 16x16: VGPR v -> M = v + 8*(lane>=16), N = lane%16
// ---------------------------------------------------------------------------
__global__ __launch_bounds__(256)
void gemm64_wmma(const float* __restrict__ X, const float* __restrict__ W,
                 const float* __restrict__ bias, float* __restrict__ Y,
                 int nrows, int do_relu)
{
    __shared__ float Wt[64 * 64];          // Wt[c*64 + k] = W[k*64 + c]
    {
        const float4* Wg4 = (const float4*)W;
#pragma unroll
        for (int i = threadIdx.x; i < 1024; i += 256) {
            const int k  = i >> 4;          // source row  (K index)
            const int c4 = (i & 15) * 4;    // source cols (N index)
            float4 w = Wg4[i];
            Wt[(c4 + 0) * 64 + k] = w.x;
            Wt[(c4 + 1) * 64 + k] = w.y;
            Wt[(c4 + 2) * 64 + k] = w.z;
            Wt[(c4 + 3) * 64 + k] = w.w;
        }
    }
    __syncthreads();

    const int wave = threadIdx.x >> 5;
    const int lane = threadIdx.x & 31;
    const int tile = blockIdx.x * 8 + wave;
    if (tile * 16 >= nrows) return;

    const int r0 = tile * 16;
    const int lm = lane & 15;
    const int hi = lane >> 4;               // half-wave: K pair {0,1} vs {2,3}

    v8f acc[4];
#pragma unroll
    for (int j = 0; j < 4; ++j)
#pragma unroll
        for (int v = 0; v < 8; ++v) acc[j][v] = 0.0f;

    const float* XrowA = X + (size_t)(r0 + lm) * 64 + 2 * hi;
    const float* WtB   = Wt + lm * 64 + 2 * hi;

#pragma unroll
    for (int k0 = 0; k0 < 64; k0 += 4) {
        v2f a = *(const v2f*)(XrowA + k0);                  // A[M=lm, K pair]
#pragma unroll
        for (int j = 0; j < 4; ++j) {
            v2f b = *(const v2f*)(WtB + j * (16 * 64) + k0); // B[K pair, N=j*16+lm]
            acc[j] = __builtin_amdgcn_wmma_f32_16x16x4_f32(
                false, a, false, b, (short)0, acc[j], false, false);
        }
    }

#pragma unroll
    for (int j = 0; j < 4; ++j) {
        const int col = j * 16 + lm;
        const float bb = bias ? bias[col] : 0.0f;
#pragma unroll
        for (int v = 0; v < 8; ++v) {
            const int row = r0 + v + 8 * hi;
            float val = acc[j][v] + bb;
            if (do_relu) val = fmaxf(val, 0.0f);
            Y[(size_t)row * 64 + col] = val;
        }
    }
}

// ---------------------------------------------------------------------------
// Elementwise / init kernels
// ---------------------------------------------------------------------------
__global__ void fill_f32(float* __restrict__ p, float v, long long n)
{
    long long i = (long long)blockIdx.x * blockDim.x + threadIdx.x;
    if (i < n) p[i] = v;
}

// deg -> 1/sqrt(deg)  (deg >= 1 always, thanks to self-loops)
__global__ void deg_to_dinv(float* __restrict__ deg, int n)
{
    int i = blockIdx.x * blockDim.x + threadIdx.x;
    if (i < n) deg[i] = rsqrtf(deg[i]);
}

// agg[n][f] = dinv[n]^2 * xw[n][f]   (self-loop term of GCN aggregation)
__global__ void agg_init(float* __restrict__ agg, const float* __restrict__ xw,
                         const float* __restrict__ dinv, long long n64)
{
    long long i = (long long)blockIdx.x * blockDim.x + threadIdx.x;
    if (i >= n64) return;
    float dv = dinv[(int)(i >> 6)];
    agg[i] = dv * dv * xw[i];
}

// ---------------------------------------------------------------------------
// Edge-parallel scatters: 32 lanes per edge, 2 features per lane.
// xw tables are N*64*4B = 25.6MB -> resident in 192MB L2; atomics resolve at L2.
// ---------------------------------------------------------------------------
__global__ void deg_count(const long long* __restrict__ dst,
                          float* __restrict__ deg, int E)
{
    int e = blockIdx.x * blockDim.x + threadIdx.x;
    if (e < E) atomicAdd(&deg[(int)dst[e]], 1.0f);
}

// residual[src[e]][f] += adj[e] * xw[dst[e]][f]
__global__ void residual_scatter(const long long* __restrict__ src,
                                 const long long* __restrict__ dst,
                                 const float* __restrict__ adj,
                                 const float* __restrict__ xw,
                                 float* __restrict__ res, int E)
{
    long long idx = (long long)blockIdx.x * blockDim.x + threadIdx.x;
    int e = (int)(idx >> 5);
    if (e >= E) return;
    int fp = ((int)idx & 31) * 2;
    int s = (int)src[e], d = (int)dst[e];
    float a = adj[e];
    const float* xv = xw + (size_t)d * 64 + fp;
    float* rv = res + (size_t)s * 64 + fp;
    atomicAdd(rv,     a * xv[0]);
    atomicAdd(rv + 1, a * xv[1]);
}

// agg[dst[e]][f] += dinv[src]*dinv[dst] * xw[src[e]][f]
__global__ void conv_scatter(const long long* __restrict__ src,
                             const long long* __restrict__ dst,
                             const float* __restrict__ dinv,
                             const float* __restrict__ xw,
                             float* __restrict__ agg, int E)
{
    long long idx = (long long)blockIdx.x * blockDim.x + threadIdx.x;
    int e = (int)(idx >> 5);
    if (e >= E) return;
    int fp = ((int)idx & 31) * 2;
    int s = (int)src[e], d = (int)dst[e];
    float w = dinv[s] * dinv[d];
    const float* xv = xw + (size_t)s * 64 + fp;
    float* av = agg + (size_t)d * 64 + fp;
    atomicAdd(av,     w * xv[0]);
    atomicAdd(av + 1, w * xv[1]);
}

// ---------------------------------------------------------------------------
// LayerNorm(64) + affine + ReLU, in place. One wave32 per node:
// lane owns features {lane, lane+32}; reductions via __shfl_xor.
// conv bias cb is added before normalization (matches reference gcn_conv + LN).
// ---------------------------------------------------------------------------
__global__ __launch_bounds__(256)
void ln_relu(float* __restrict__ h, const float* __restrict__ cb,
             const float* __restrict__ g, const float* __restrict__ bln, int n)
{
    const int wave = threadIdx.x >> 5;
    const int lane = threadIdx.x & 31;
    const int node = blockIdx.x * 8 + wave;
    if (node >= n) return;

    float* row = h + (size_t)node * 64;
    float v0 = row[lane]      + cb[lane];
    float v1 = row[lane + 32] + cb[lane + 32];

    float s = v0 + v1;
#pragma unroll
    for (int m = 16; m >= 1; m >>= 1) s += __shfl_xor(s, m, 32);
    float mu = s * 0.015625f;                    // /64

    float d0 = v0 - mu, d1 = v1 - mu;
    float q = d0 * d0 + d1 * d1;
#pragma unroll
    for (int m = 16; m >= 1; m >>= 1) q += __shfl_xor(q, m, 32);
    float rinv = rsqrtf(q * 0.015625f + 1e-5f);

    row[lane]      = fmaxf(0.0f, d0 * rinv * g[lane]      + bln[lane]);
    row[lane + 32] = fmaxf(0.0f, d1 * rinv * g[lane + 32] + bln[lane + 32]);
}

// ---------------------------------------------------------------------------
// Orchestration (all launches on `stream`; graph-capture safe).
// ---------------------------------------------------------------------------
extern "C" void kernel_launch(void* const* d_in, const int* in_sizes, int n_in,
                              void* d_out, int out_size, void* d_ws, size_t ws_size,
                              hipStream_t stream)
{
    const float*     x     = (const float*)d_in[0];      // [N,64]
    const float*     x_org = (const float*)d_in[1];      // [N,64]
    const float*     adj   = (const float*)d_in[2];      // [E]
    const float*     Wi    = (const float*)d_in[3];      // [64,64]
    const float*     bi    = (const float*)d_in[4];      // [64]
    const float*     convW = (const float*)d_in[5];      // [L,64,64]
    const float*     convB = (const float*)d_in[6];      // [L,64]
    const float*     lnG   = (const float*)d_in[7];      // [L,64]
    const float*     lnB   = (const float*)d_in[8];      // [L,64]
    const float*     Wlin  = (const float*)d_in[9];      // [64,64]
    const float*     blin  = (const float*)d_in[10];     // [64]
    const float*     Wres  = (const float*)d_in[11];     // [64,64]
    const long long* eidx  = (const long long*)d_in[12]; // [2,E] int64

    const int N = in_sizes[0] / 64;
    const int E = in_sizes[2];
    const int L = in_sizes[5] / (64 * 64);
    const long long N64 = (long long)N * 64;

    const long long* srcI = eidx;       // edge_index[0]
    const long long* dstI = eidx + E;   // edge_index[1]

    float* out = (float*)d_out;         // output 0: [N,64]
    float* res = out + N64;             // output 1: residual [N,64]

    // workspace: hA | hB | xw | deg
    float* hA  = (float*)d_ws;
    float* hB  = hA + N64;
    float* xw  = hB + N64;
    float* deg = xw + N64;

    const int gemmBlocks  = ((N + 15) / 16 + 7) / 8;
    const int elemBlocks  = (int)((N64 + 255) / 256);
    const int edgeFBlocks = (int)(((long long)E * 32 + 255) / 256);
    const int edgeBlocks  = (E + 255) / 256;
    const int nodeBlocks  = (N + 255) / 256;
    const int lnBlocks    = (N + 7) / 8;

    // ---- residual branch: res = scatter_add(adj * (x_org @ Wres)[dst] -> src)
    gemm64_wmma<<<gemmBlocks, 256, 0, stream>>>(x_org, Wres, nullptr, xw, N, 0);
    fill_f32<<<elemBlocks, 256, 0, stream>>>(res, 0.0f, N64);
    residual_scatter<<<edgeFBlocks, 256, 0, stream>>>(srcI, dstI, adj, xw, res, E);

    // ---- degrees with self-loops: deg[n] = 1 + #(dst == n); dinv = rsqrt(deg)
    fill_f32<<<nodeBlocks, 256, 0, stream>>>(deg, 1.0f, N);
    deg_count<<<edgeBlocks, 256, 0, stream>>>(dstI, deg, E);
    deg_to_dinv<<<nodeBlocks, 256, 0, stream>>>(deg, N);   // deg now holds dinv

    // ---- h0 = relu(x @ Wi + bi)
    gemm64_wmma<<<gemmBlocks, 256, 0, stream>>>(x, Wi, bi, hA, N, 1);

    // ---- L x { xw = h@W ; agg = D^-1/2 A D^-1/2 xw ; h = relu(LN(agg + b)) }
    float* cur = hA;
    float* nxt = hB;
    for (int l = 0; l < L; ++l) {
        gemm64_wmma<<<gemmBlocks, 256, 0, stream>>>(cur, convW + (size_t)l * 64 * 64,
                                                    nullptr, xw, N, 0);
        agg_init<<<elemBlocks, 256, 0, stream>>>(nxt, xw, deg, N64);
        conv_scatter<<<edgeFBlocks, 256, 0, stream>>>(srcI, dstI, deg, xw, nxt, E);
        ln_relu<<<lnBlocks, 256, 0, stream>>>(nxt, convB + (size_t)l * 64,
                                              lnG + (size_t)l * 64,
                                              lnB + (size_t)l * 64, N);
        float* t = cur; cur = nxt; nxt = t;
    }

    // ---- out = h @ Wl + bl
    gemm64_wmma<<<gemmBlocks, 256, 0, stream>>>(cur, Wlin, blin, out, N, 0);
}